// PoseHead_42219528520129
// MI455X (gfx1250) — compile-verified
//
#include <hip/hip_runtime.h>
#include <math.h>

// Problem constants (from reference)
#define NCH   64      // C: channels
#define NSEG  64      // B: segments
#define NHID  256     // HID
#define NQD   4
#define NTD   3

typedef __attribute__((ext_vector_type(2))) float v2f;
typedef __attribute__((ext_vector_type(4))) float v4f;
typedef __attribute__((ext_vector_type(8))) float v8f;

// ---------------------------------------------------------------------------
// Kernel 0: zero the accumulator region of the workspace (harness poisons ws)
// ---------------------------------------------------------------------------
__global__ void zero_ws_kernel(float* __restrict__ ws, int n) {
    int i = blockIdx.x * blockDim.x + threadIdx.x;
    if (i < n) ws[i] = 0.0f;
}

// ---------------------------------------------------------------------------
// Kernel 1: single streaming pass over mask+prediction (HBM-bound, ~258 MB).
// Per (segment, channel): accumulate sum(exp(mask)) and sum(exp(mask)*pred).
// Mathematically identical to softmax-then-weighted-sum (exp(max) cancels;
// mask ~ N(0,1) so fp32 exp never overflows).
//
// 256 threads/block: 16 channel-quads x 16 row-lanes. Each thread streams
// float4 (global_load_b128, nontemporal: read-once stream must not evict the
// resident accumulators/weights from L2). Sorted segment ids -> register
// accumulation with flush-on-segment-change into 32KB LDS accumulators,
// then sparse global fp32 atomics.
// ---------------------------------------------------------------------------
__global__ void seg_pass_kernel(const float* __restrict__ pred,
                                const float* __restrict__ mask,
                                const int*   __restrict__ seg,
                                float* __restrict__ g_sum,   // [NSEG*NCH]
                                float* __restrict__ g_wp,    // [NSEG*NCH]
                                int nRows, int rowsPerBlock) {
    __shared__ float s_sum[NSEG * NCH];
    __shared__ float s_wp [NSEG * NCH];
    const int tid = threadIdx.x;
    for (int i = tid; i < NSEG * NCH; i += 256) { s_sum[i] = 0.0f; s_wp[i] = 0.0f; }
    __syncthreads();

    const int cq = (tid & 15) * 4;   // channel quad base: 0,4,...,60
    const int rs = tid >> 4;         // row sub-lane 0..15
    const int r0 = blockIdx.x * rowsPerBlock;
    const int r1 = min(r0 + rowsPerBlock, nRows);

    int curSeg = -1;
    v4f accE = {}, accP = {};

    for (int r = r0 + rs; r < r1; r += 16) {
        size_t idx = (size_t)r * NCH + cq;
        if (r + 64 < r1) {  // gfx1250 global_prefetch_b8, 4 iterations ahead
            __builtin_prefetch(mask + idx + 64 * NCH, 0, 1);
            __builtin_prefetch(pred + idx + 64 * NCH, 0, 1);
        }
        int s = seg[r];
        v4f m = __builtin_nontemporal_load((const v4f*)(mask + idx));
        v4f p = __builtin_nontemporal_load((const v4f*)(pred + idx));
        v4f e;
        e.x = __expf(m.x); e.y = __expf(m.y); e.z = __expf(m.z); e.w = __expf(m.w);
        if (s != curSeg) {
            if (curSeg >= 0) {
                float* bs = &s_sum[curSeg * NCH + cq];
                float* bw = &s_wp [curSeg * NCH + cq];
                atomicAdd(bs + 0, accE.x); atomicAdd(bw + 0, accP.x);
                atomicAdd(bs + 1, accE.y); atomicAdd(bw + 1, accP.y);
                atomicAdd(bs + 2, accE.z); atomicAdd(bw + 2, accP.z);
                atomicAdd(bs + 3, accE.w); atomicAdd(bw + 3, accP.w);
            }
            curSeg = s;
            accE = (v4f){};
            accP = (v4f){};
        }
        accE += e;
        accP += e * p;
    }
    if (curSeg >= 0) {
        float* bs = &s_sum[curSeg * NCH + cq];
        float* bw = &s_wp [curSeg * NCH + cq];
        atomicAdd(bs + 0, accE.x); atomicAdd(bw + 0, accP.x);
        atomicAdd(bs + 1, accE.y); atomicAdd(bw + 1, accP.y);
        atomicAdd(bs + 2, accE.z); atomicAdd(bw + 2, accP.z);
        atomicAdd(bs + 3, accE.w); atomicAdd(bw + 3, accP.w);
    }
    __syncthreads();

    // Sparse flush: only segments this block actually touched.
    for (int i = tid; i < NSEG * NCH; i += 256) {
        float se = s_sum[i];
        if (se != 0.0f) {
            unsafeAtomicAdd(&g_sum[i], se);       // global_atomic_add_f32
            unsafeAtomicAdd(&g_wp [i], s_wp[i]);
        }
    }
}

// ---------------------------------------------------------------------------
// Kernel 2: gp = g_wp / g_sum  (guard empty segments -> 0, matching the
// reference's segment_sum semantics), then
//   hidden[64,256] = gp[64,64] @ W_h^T[64,256] + b_h     via fp32 WMMA.
// V_WMMA_F32_16X16X4_F32 tiles: 4 M-tiles x 16 N-tiles = 64 tiles,
// 8 waves -> 8 tiles/wave, 16 K-steps of 4 each.
// VGPR layouts per ISA 7.12.2:
//   A (16x4):  lane<16 -> M=lane,K=v ; lane>=16 -> M=lane-16,K=2+v
//   B (4x16):  mirrored: N=lane%16, K=v+2*(lane>=16)
//   C/D (16x16): VGPR i, lane l -> M=i+8*(l>=16), N=l%16
// ---------------------------------------------------------------------------
#define SGP_STRIDE 65   // pad to avoid LDS bank conflicts (stride%64==1)

__global__ void mlp_wmma_kernel(const float* __restrict__ g_sum,
                                const float* __restrict__ g_wp,
                                const float* __restrict__ W_h,   // [256][64]
                                const float* __restrict__ b_h,   // [256]
                                float* __restrict__ hid) {       // [64][256]
    __shared__ float s_gp[NSEG * SGP_STRIDE];
    const int tid = threadIdx.x;

    for (int i = tid; i < NSEG * NCH; i += 256) {
        int b = i >> 6, c = i & 63;
        float se = g_sum[i];
        s_gp[b * SGP_STRIDE + c] = (se > 0.0f) ? (g_wp[i] / se) : 0.0f;
    }
    __syncthreads();

    const int lane = tid & 31;
    const int wave = tid >> 5;
    const int row  = lane & 15;
    const int hi   = lane >> 4;

    for (int tile = wave; tile < 64; tile += 8) {
        const int m0 = (tile & 3) * 16;
        const int n0 = (tile >> 2) * 16;
        v8f acc = {};
#pragma unroll
        for (int k0 = 0; k0 < NCH; k0 += 4) {
            const int kb = k0 + 2 * hi;
            v2f a, b;
            a.x = s_gp[(m0 + row) * SGP_STRIDE + kb];
            a.y = s_gp[(m0 + row) * SGP_STRIDE + kb + 1];
            const float* wrow = W_h + (size_t)(n0 + row) * NCH + kb;  // B[k][n]=W_h[n][k]
            b.x = wrow[0];
            b.y = wrow[1];
            acc = __builtin_amdgcn_wmma_f32_16x16x4_f32(
                /*neg_a=*/false, a, /*neg_b=*/false, b,
                /*c_mod=*/(short)0, acc, /*reuse_a=*/false, /*reuse_b=*/false);
        }
        const float bias = b_h[n0 + row];
#pragma unroll
        for (int i = 0; i < 8; ++i) {
            int m = m0 + i + 8 * hi;
            hid[(size_t)m * NHID + n0 + row] = acc[i] + bias;
        }
    }
}

// ---------------------------------------------------------------------------
// Kernel 3: heads. q = hidden@W_q^T + b_q (then L2-normalize exactly as the
// reference), t = hidden@W_t^T + b_t. One thread per batch row.
// Output: q[64,4] then t[64,3], flat.
// ---------------------------------------------------------------------------
__global__ void heads_kernel(const float* __restrict__ hid,  // [64][256]
                             const float* __restrict__ W_q,  // [4][256]
                             const float* __restrict__ b_q,  // [4]
                             const float* __restrict__ W_t,  // [3][256]
                             const float* __restrict__ b_t,  // [3]
                             float* __restrict__ out) {      // [448]
    const int b = threadIdx.x;
    if (b >= NSEG) return;
    float q[NQD], t[NTD];
#pragma unroll
    for (int j = 0; j < NQD; ++j) q[j] = b_q[j];
#pragma unroll
    for (int j = 0; j < NTD; ++j) t[j] = b_t[j];

    for (int h = 0; h < NHID; ++h) {
        float hv = hid[(size_t)b * NHID + h];
#pragma unroll
        for (int j = 0; j < NQD; ++j) q[j] = fmaf(hv, W_q[j * NHID + h], q[j]);
#pragma unroll
        for (int j = 0; j < NTD; ++j) t[j] = fmaf(hv, W_t[j * NHID + h], t[j]);
    }
    float ss = 1e-10f;
#pragma unroll
    for (int j = 0; j < NQD; ++j) ss += q[j] * q[j];
    float inv = 1.0f / (sqrtf(ss) + 1e-10f);
#pragma unroll
    for (int j = 0; j < NQD; ++j) out[b * NQD + j] = q[j] * inv;
#pragma unroll
    for (int j = 0; j < NTD; ++j) out[NSEG * NQD + b * NTD + j] = t[j];
}

// ---------------------------------------------------------------------------
extern "C" void kernel_launch(void* const* d_in, const int* in_sizes, int n_in,
                              void* d_out, int out_size, void* d_ws, size_t ws_size,
                              hipStream_t stream) {
    const float* pred = (const float*)d_in[0];
    const float* mask = (const float*)d_in[1];
    const int*   seg  = (const int*)  d_in[2];   // jnp int64 w/o x64 -> int32
    const float* W_h  = (const float*)d_in[3];
    const float* b_h  = (const float*)d_in[4];
    const float* W_q  = (const float*)d_in[5];
    const float* b_q  = (const float*)d_in[6];
    const float* W_t  = (const float*)d_in[7];
    const float* b_t  = (const float*)d_in[8];
    float* out = (float*)d_out;

    const int nRows = in_sizes[0] / NCH;  // 524288

    // Workspace: [0,4096) sum(exp), [4096,8192) sum(exp*pred), [8192,24576) hidden
    float* ws    = (float*)d_ws;
    float* g_sum = ws;
    float* g_wp  = ws + NSEG * NCH;
    float* hid   = ws + 2 * NSEG * NCH;

    // 0) zero accumulators
    zero_ws_kernel<<<(2 * NSEG * NCH + 255) / 256, 256, 0, stream>>>(ws, 2 * NSEG * NCH);

    // 1) streaming segment reduction (bandwidth-bound: ~258 MB @ 23.3 TB/s)
    const int rowsPerBlock = 2048;
    const int nBlocks = (nRows + rowsPerBlock - 1) / rowsPerBlock;
    seg_pass_kernel<<<nBlocks, 256, 0, stream>>>(pred, mask, seg, g_sum, g_wp,
                                                 nRows, rowsPerBlock);

    // 2) gp -> hidden via fp32 WMMA (one block, 8 waves)
    mlp_wmma_kernel<<<1, 256, 0, stream>>>(g_sum, g_wp, W_h, b_h, hid);

    // 3) heads + q normalization
    heads_kernel<<<1, 64, 0, stream>>>(hid, W_q, b_q, W_t, b_t, out);
}